// MTFE_39779987095798
// MI455X (gfx1250) — compile-verified
//
#include <hip/hip_runtime.h>
#include <math.h>

// Problem constants (fixed by reference: B=4, H=W=512, C_p=64)
constexpr int IMG_H  = 512;
constexpr int IMG_W  = 512;
constexpr int IMG_HW = IMG_H * IMG_W;       // 262144
constexpr int NB     = 4;
constexpr int NPIX   = NB * IMG_HW;         // 1048576

// Filter table layout (floats) inside workspace filter region
constexpr int FO_G7   = 0;     // gauss 7x7  sigma=1.0
constexpr int FO_G13  = 49;    // gauss 13x13 sigma=2.0
constexpr int FO_B5   = 218;   // DoG 5x5  s=0.5
constexpr int FO_B7   = 243;   // DoG 7x7  s=1.0
constexpr int FO_B11  = 292;   // DoG 11x11 s=1.5
constexpr int FO_B13  = 413;   // DoG 13x13 s=2.0
constexpr int FILT_TOTAL = 582;

typedef float v2f __attribute__((ext_vector_type(2)));
typedef float v8f __attribute__((ext_vector_type(8)));

// ---------------------------------------------------------------------------
// Precompute: folded attention coeffs (3x9), folded output matrix (64x12),
// and all Gaussian/DoG filter taps — all built on-device (graph-capture safe).
// ---------------------------------------------------------------------------
__device__ void d_gauss(int ks, float sigma, float* out) {
  int h = ks / 2;
  float sum = 0.f;
  for (int i = 0; i < ks * ks; ++i) {
    float yy = (float)(i / ks - h), xx = (float)(i % ks - h);
    float v = expf(-(xx * xx + yy * yy) / (2.f * sigma * sigma));
    out[i] = v; sum += v;
  }
  float inv = 1.f / sum;
  for (int i = 0; i < ks * ks; ++i) out[i] *= inv;
}

__device__ void d_dog(int ks, float s, float* out) {
  int h = ks / 2;
  float s1 = 1.6f * s, s2 = s;
  float sumA = 0.f, sumB = 0.f;
  for (int i = 0; i < ks * ks; ++i) {
    float yy = (float)(i / ks - h), xx = (float)(i % ks - h);
    float r2 = xx * xx + yy * yy;
    sumA += expf(-r2 / (2.f * s1 * s1));
    sumB += expf(-r2 / (2.f * s2 * s2));
  }
  float ia = 1.f / sumA, ib = 1.f / sumB;
  for (int i = 0; i < ks * ks; ++i) {
    float yy = (float)(i / ks - h), xx = (float)(i % ks - h);
    float r2 = xx * xx + yy * yy;
    out[i] = expf(-r2 / (2.f * s1 * s1)) * ia - expf(-r2 / (2.f * s2 * s2)) * ib;
  }
}

__global__ __launch_bounds__(256) void mtfe_precompute(
    const float* __restrict__ w_tex,  const float* __restrict__ b_tex,
    const float* __restrict__ w_diff, const float* __restrict__ b_diff,
    const float* __restrict__ w_band, const float* __restrict__ b_band,
    const float* __restrict__ w_attn, const float* __restrict__ b_attn,
    float* __restrict__ attnC,  // 27 floats: 3 rows x (8 coeffs + bias)
    float* __restrict__ Wout,   // 64x12 row-major
    float* __restrict__ filt)   // FILT_TOTAL floats
{
  int tid = threadIdx.x;
  // --- folded attention logits: A_k = sum_j attnC[k][j]*f_j + attnC[k][8]
  if (tid < 27) {
    int k = tid / 9, j = tid % 9;
    const float* wa = w_attn + k * 192;
    float acc = 0.f;
    if (j < 2)       { for (int o = 0; o < 64; ++o) acc += wa[o]       * w_tex [o * 2 + j]; }
    else if (j < 4)  { for (int o = 0; o < 64; ++o) acc += wa[64 + o]  * w_diff[o * 2 + (j - 2)]; }
    else if (j < 8)  { for (int o = 0; o < 64; ++o) acc += wa[128 + o] * w_band[o * 4 + (j - 4)]; }
    else {
      for (int o = 0; o < 64; ++o)
        acc += wa[o] * b_tex[o] + wa[64 + o] * b_diff[o] + wa[128 + o] * b_band[o];
      acc += b_attn[k];
    }
    attnC[k * 9 + j] = acc;
  }
  // --- filter taps (one thread per filter, serial; trivial work)
  if (tid == 32) d_gauss(7,  1.0f, filt + FO_G7);
  if (tid == 33) d_gauss(13, 2.0f, filt + FO_G13);
  if (tid == 34) d_dog(5,  0.5f, filt + FO_B5);
  if (tid == 35) d_dog(7,  1.0f, filt + FO_B7);
  if (tid == 36) d_dog(11, 1.5f, filt + FO_B11);
  if (tid == 37) d_dog(13, 2.0f, filt + FO_B13);
  // --- folded output matrix: out[o] = sum_k Wout[o][k] * e[k]
  for (int idx = tid; idx < 64 * 12; idx += 256) {
    int o = idx / 12, j = idx % 12;
    float v;
    if (j < 2)       v = w_tex [o * 2 + j];
    else if (j < 4)  v = w_diff[o * 2 + (j - 2)];
    else if (j < 8)  v = w_band[o * 4 + (j - 4)];
    else if (j == 8) v = b_tex[o];
    else if (j == 9) v = b_diff[o];
    else if (j == 10) v = b_band[o];
    else             v = 0.f;
    Wout[idx] = v;
  }
}

// ---------------------------------------------------------------------------
// Feature kernel: LDS-tiled convs. Writes planes 0,1,3,4,5,6,7 of feat[8][NPIX].
// Zero padding (lax.conv default). Max radius 6 -> 28x28 LDS tile.
// ---------------------------------------------------------------------------
__device__ __forceinline__ float conv_sm(const float sm[28][28], int ly, int lx,
                                         const float* __restrict__ k, int r) {
  float acc = 0.f;
  int ks = 2 * r + 1;
  for (int dy = -r; dy <= r; ++dy)
    for (int dx = -r; dx <= r; ++dx)
      acc = fmaf(k[(dy + r) * ks + (dx + r)], sm[ly + dy][lx + dx], acc);
  return acc;
}

__global__ __launch_bounds__(256) void mtfe_features(
    const float* __restrict__ I, float* __restrict__ feat,
    const float* __restrict__ filt)
{
  constexpr int R = 6, T = 16, TT = T + 2 * R;  // 28
  __shared__ float sm[TT][TT];
  __shared__ float sf[FILT_TOTAL];
  int tx = threadIdx.x, ty = threadIdx.y;
  int tid = ty * T + tx;
  int bx = blockIdx.x * T, by = blockIdx.y * T, bb = blockIdx.z;
  const float* img = I + (size_t)bb * IMG_HW;
  for (int i = tid; i < FILT_TOTAL; i += 256) sf[i] = filt[i];
  for (int i = tid; i < TT * TT; i += 256) {
    int yy = by + i / TT - R, xx = bx + (i % TT) - R;
    sm[i / TT][i % TT] =
        (yy >= 0 && yy < IMG_H && xx >= 0 && xx < IMG_W) ? img[yy * IMG_W + xx] : 0.f;
  }
  __syncthreads();
  int ly = ty + R, lx = tx + R;
  // Sobel + Laplacian (3x3)
  float tl = sm[ly-1][lx-1], tc = sm[ly-1][lx], tr = sm[ly-1][lx+1];
  float ml = sm[ly  ][lx-1], mc = sm[ly  ][lx], mr = sm[ly  ][lx+1];
  float bl = sm[ly+1][lx-1], bc = sm[ly+1][lx], br = sm[ly+1][lx+1];
  float gx = (tr - tl) + 2.f * (mr - ml) + (br - bl);
  float gy = (bl - tl) + 2.f * (bc - tc) + (br - tr);
  float G = sqrtf(gx * gx + gy * gy + 1e-8f);
  float L = fabsf(tc + ml + mr + bc - 4.f * mc);
  // Gaussians + DoG bands
  float gs1 = conv_sm(sm, ly, lx, sf + FO_G7, 3);
  float gs2 = conv_sm(sm, ly, lx, sf + FO_G13, 6);
  float c5  = conv_sm(sm, ly, lx, sf + FO_B5, 2);
  float c7  = conv_sm(sm, ly, lx, sf + FO_B7, 3);
  float c11 = conv_sm(sm, ly, lx, sf + FO_B11, 5);
  float c13 = conv_sm(sm, ly, lx, sf + FO_B13, 6);

  size_t p = (size_t)bb * IMG_HW + (size_t)(by + ty) * IMG_W + (bx + tx);
  feat[0 * (size_t)NPIX + p] = G;
  feat[1 * (size_t)NPIX + p] = L;
  // plane 2 (D) written by last diffusion iteration
  feat[3 * (size_t)NPIX + p] = fabsf(gs2 - gs1);
  feat[4 * (size_t)NPIX + p] = fabsf(c5);
  feat[5 * (size_t)NPIX + p] = fabsf(c7);
  feat[6 * (size_t)NPIX + p] = fabsf(c11);
  feat[7 * (size_t)NPIX + p] = fabsf(c13);
}

// ---------------------------------------------------------------------------
// One Perona-Malik iteration. Edge-clamped gradients, zero-padded divergence.
// Image is 4 MB -> fully L2 resident; direct loads are fine.
// ---------------------------------------------------------------------------
__device__ __forceinline__ float d_atc(const float* __restrict__ s, int y, int x) {
  y = min(max(y, 0), IMG_H - 1);
  x = min(max(x, 0), IMG_W - 1);
  return s[y * IMG_W + x];
}
__device__ __forceinline__ void d_cg(const float* __restrict__ s, int y, int x,
                                     float& cgx, float& cgy) {
  float gx = d_atc(s, y, x + 1) - d_atc(s, y, x - 1);
  float gy = d_atc(s, y + 1, x) - d_atc(s, y - 1, x);
  float gm = sqrtf(gx * gx + gy * gy + 1e-8f);
  float c = expf(-gm * gm);  // kappa = 1
  cgx = c * gx; cgy = c * gy;
}

__global__ __launch_bounds__(256) void mtfe_diffuse(const float* __restrict__ src,
                                                    float* __restrict__ dst)
{
  int idx = blockIdx.x * 256 + threadIdx.x;
  if (idx >= NPIX) return;
  int bb = idx / IMG_HW, rem = idx % IMG_HW;
  int y = rem / IMG_W, x = rem % IMG_W;
  const float* s = src + (size_t)bb * IMG_HW;
  float t1 = 0.f, t2 = 0.f, t3 = 0.f, t4 = 0.f, dmy;
  if (x + 1 < IMG_W) d_cg(s, y, x + 1, t1, dmy);
  if (x - 1 >= 0)    d_cg(s, y, x - 1, t2, dmy);
  if (y + 1 < IMG_H) d_cg(s, y + 1, x, dmy, t3);
  if (y - 1 >= 0)    d_cg(s, y - 1, x, dmy, t4);
  dst[idx] = s[rem] + 0.1f * ((t1 - t2) + (t3 - t4));
}

// ---------------------------------------------------------------------------
// Fusion: per-pixel softmax attention, then 64x12 GEMM via
// V_WMMA_F32_16X16X4_F32 (4 M-tiles x 2 N-tiles x 3 K-steps per 32 pixels).
// A (16x4 f32): lane M=lane&15; VGPR0 K = (lane>=16 ? 2 : 0)+kb*4; VGPR1 K+1.
// B (4x16 f32): lane N=lane&15; same K split.
// D (16x16 f32): VGPR i -> M = i + 8*(lane>=16), N = lane&15.
// ---------------------------------------------------------------------------
__global__ __launch_bounds__(256) void mtfe_fuse(
    const float* __restrict__ feat, const float* __restrict__ attnC,
    const float* __restrict__ Wout, float* __restrict__ out)
{
  __shared__ float eL[8][32 * 13];  // per-wave 32 pixels x 12 (stride 13)
  int lane = threadIdx.x & 31;
  int wv   = threadIdx.x >> 5;
  int wgid = blockIdx.x * 8 + wv;
  int nw   = gridDim.x * 8;
  const int ngroups = NPIX / 32;  // 32768; divisible by nw for our grid

  // attention coefficients (wave-uniform scalar loads)
  float C[27];
#pragma unroll
  for (int i = 0; i < 27; ++i) C[i] = attnC[i];

  // A-matrix fragments, preloaded once per wave
  int mrow = lane & 15;
  int khalf = (lane >> 4) * 2;
  v2f afrag[4][3];
#pragma unroll
  for (int m = 0; m < 4; ++m)
#pragma unroll
    for (int kb = 0; kb < 3; ++kb) {
      int o = m * 16 + mrow, k = kb * 4 + khalf;
      afrag[m][kb].x = Wout[o * 12 + k];
      afrag[m][kb].y = Wout[o * 12 + k + 1];
    }

  for (int g = wgid; g < ngroups; g += nw) {
    int p = g * 32 + lane;
    float f[8];
#pragma unroll
    for (int j = 0; j < 8; ++j) f[j] = feat[(size_t)j * NPIX + p];

    float a0 = C[8], a1 = C[17], a2 = C[26];
#pragma unroll
    for (int j = 0; j < 8; ++j) {
      a0 = fmaf(C[j], f[j], a0);
      a1 = fmaf(C[9 + j], f[j], a1);
      a2 = fmaf(C[18 + j], f[j], a2);
    }
    float mx = fmaxf(a0, fmaxf(a1, a2));
    float x0 = expf(a0 - mx), x1 = expf(a1 - mx), x2 = expf(a2 - mx);
    float inv = 1.f / (x0 + x1 + x2);
    float w0 = x0 * inv, w1 = x1 * inv, w2 = x2 * inv;

    float ev[12] = {w0 * f[0], w0 * f[1], w1 * f[2], w1 * f[3],
                    w2 * f[4], w2 * f[5], w2 * f[6], w2 * f[7],
                    w0, w1, w2, 0.f};
    float* el = eL[wv];
#pragma unroll
    for (int k = 0; k < 12; ++k) el[lane * 13 + k] = ev[k];
    __syncthreads();

    int b = p / IMG_HW;               // 32-pixel groups never cross batches
    int hwBase = (g * 32) % IMG_HW;

#pragma unroll
    for (int t = 0; t < 2; ++t) {
      int n = (lane & 15) + 16 * t;
      v2f bfrag[3];
#pragma unroll
      for (int kb = 0; kb < 3; ++kb) {
        bfrag[kb].x = el[n * 13 + kb * 4 + khalf];
        bfrag[kb].y = el[n * 13 + kb * 4 + khalf + 1];
      }
#pragma unroll
      for (int m = 0; m < 4; ++m) {
        int hw = hwBase + 16 * t + (lane & 15);
        int obase = m * 16 + (lane >> 4) * 8;
        float* op = out + ((size_t)b * 64 + obase) * IMG_HW + hw;
#if __has_builtin(__builtin_amdgcn_wmma_f32_16x16x4_f32)
        v8f acc = {0.f, 0.f, 0.f, 0.f, 0.f, 0.f, 0.f, 0.f};
#pragma unroll
        for (int kb = 0; kb < 3; ++kb)
          acc = __builtin_amdgcn_wmma_f32_16x16x4_f32(
              false, afrag[m][kb], false, bfrag[kb], (short)0, acc, false, false);
#pragma unroll
        for (int i = 0; i < 8; ++i) op[(size_t)i * IMG_HW] = acc[i];
#else
        // scalar fallback (keeps compile alive if builtin absent)
        int nn = 16 * t + (lane & 15);
#pragma unroll
        for (int i = 0; i < 8; ++i) {
          int o = obase + i;
          float s = 0.f;
          for (int k = 0; k < 12; ++k) s = fmaf(Wout[o * 12 + k], el[nn * 13 + k], s);
          op[(size_t)i * IMG_HW] = s;
        }
#endif
      }
    }
    __syncthreads();
  }
}

// ---------------------------------------------------------------------------
// Host launcher. Workspace layout (floats):
//   [0, 8*NPIX)          : 8 feature planes (plane 2 = diffused D)
//   [8*NPIX, 9*NPIX)     : diffusion ping
//   [9*NPIX, 10*NPIX)    : diffusion pong
//   [10*NPIX, +32)       : attnC (27, padded)
//   [+32, +32+768)       : Wout 64x12
//   [+800, +800+582)     : filter taps
// Total ~42 MB.
// ---------------------------------------------------------------------------
extern "C" void kernel_launch(void* const* d_in, const int* in_sizes, int n_in,
                              void* d_out, int out_size, void* d_ws, size_t ws_size,
                              hipStream_t stream) {
  (void)in_sizes; (void)n_in; (void)out_size; (void)ws_size;
  const float* I_T    = (const float*)d_in[0];
  const float* w_tex  = (const float*)d_in[1];
  const float* b_tex  = (const float*)d_in[2];
  const float* w_diff = (const float*)d_in[3];
  const float* b_diff = (const float*)d_in[4];
  const float* w_band = (const float*)d_in[5];
  const float* b_band = (const float*)d_in[6];
  const float* w_attn = (const float*)d_in[7];
  const float* b_attn = (const float*)d_in[8];
  float* out = (float*)d_out;

  float* ws    = (float*)d_ws;
  float* feat  = ws;
  float* tmpA  = ws + 8 * (size_t)NPIX;
  float* tmpB  = ws + 9 * (size_t)NPIX;
  float* attnC = ws + 10 * (size_t)NPIX;
  float* Wout  = attnC + 32;
  float* filt  = attnC + 32 + 768;

  mtfe_precompute<<<1, 256, 0, stream>>>(w_tex, b_tex, w_diff, b_diff,
                                         w_band, b_band, w_attn, b_attn,
                                         attnC, Wout, filt);

  mtfe_features<<<dim3(IMG_W / 16, IMG_H / 16, NB), dim3(16, 16), 0, stream>>>(
      I_T, feat, filt);

  // 10 diffusion iterations; final iteration writes feature plane 2
  const float* src = I_T;
  for (int it = 0; it < 10; ++it) {
    float* dst = (it == 9) ? (feat + 2 * (size_t)NPIX)
                           : ((it % 2 == 0) ? tmpA : tmpB);
    mtfe_diffuse<<<NPIX / 256, 256, 0, stream>>>(src, dst);
    src = dst;
  }

  mtfe_fuse<<<1024, 256, 0, stream>>>(feat, attnC, Wout, out);
}